// NeuralUniLasso_26061861552582
// MI455X (gfx1250) — compile-verified
//
#include <hip/hip_runtime.h>
#include <math.h>

typedef __attribute__((ext_vector_type(2))) float v2f;
typedef __attribute__((ext_vector_type(8))) float v8f;

#define Bn 32768
#define Fn 512
#define Hn 10
#define PR 20   // floats per feature-pair in LDS: 5 chunks of K=4, zero padding waste

// gfx1250 hardware TRANS tanh (v_tanh_f32); exp2/rcp fallback saturates correctly.
__device__ __forceinline__ float fast_tanh(float v) {
#if __has_builtin(__builtin_amdgcn_tanhf)
  return __builtin_amdgcn_tanhf(v);
#else
  float e = __builtin_amdgcn_exp2f(v * 2.885390081777927f); // 2*log2(e)
  return 1.0f - 2.0f * __builtin_amdgcn_rcpf(e + 1.0f);
#endif
}

// Prep: theta_s = softplus(theta) -> d_out tail + ws; const0 = bias + dot(theta_s, b2) -> ws.
__global__ void __launch_bounds__(Fn) nul_prep(
    const float* __restrict__ theta, const float* __restrict__ b2,
    const float* __restrict__ bias, float* __restrict__ ws_theta,
    float* __restrict__ ws_const, float* __restrict__ out_theta) {
  __shared__ float red[Fn];
  int t = threadIdx.x;
  float ts = logf(1.0f + expf(theta[t]));
  out_theta[t] = ts;
  ws_theta[t]  = ts;
  red[t] = ts * b2[t];
  __syncthreads();
  for (int s = Fn / 2; s > 0; s >>= 1) {
    if (t < s) red[t] += red[t + s];
    __syncthreads();
  }
  if (t == 0) ws_const[0] = red[0] + bias[0];
}

// Main: each wave owns 16 rows. Feature pair (f0,f1) -> 5 WMMA K=4 chunks:
//   ch0: f0 h0-3 | ch1: f0 h4-7 | ch2: f1 h0-3 | ch3: f1 h4-7 | ch4: {f0h8,f0h9,f1h8,f1h9}
// Lane (m = lane&15, t2 = lane>>4) supplies A[m][K=2*t2+{0,1}] and the matching
// broadcast B values, so every column of D accumulates the full per-row sum.
__global__ void __launch_bounds__(128) nul_main(
    const float* __restrict__ x,  const float* __restrict__ W1,
    const float* __restrict__ b1, const float* __restrict__ W2,
    const float* __restrict__ ws_theta, const float* __restrict__ ws_const,
    float* __restrict__ y) {
  __shared__ float sW1[(Fn / 2) * PR];
  __shared__ float sB1[(Fn / 2) * PR];
  __shared__ float sC [(Fn / 2) * PR];

  int tid = threadIdx.x;
  for (int idx = tid; idx < (Fn / 2) * PR; idx += 128) {
    int p = idx / PR, r = idx % PR;
    int f, h;
    if (r < 8)       { f = 2 * p;                    h = r;                }
    else if (r < 16) { f = 2 * p + 1;                h = r - 8;            }
    else             { f = 2 * p + ((r - 16) >> 1);  h = 8 + ((r - 16) & 1); }
    sW1[idx] = W1[f * Hn + h];
    sB1[idx] = b1[f * Hn + h];
    sC [idx] = ws_theta[f] * W2[f * Hn + h];
  }
  __syncthreads();

  float c0  = ws_const[0];
  int lane  = tid & 31;
  int wave  = tid >> 5;
  int group = blockIdx.x * 4 + wave;   // 512 blocks * 4 waves = 2048 groups of 16 rows
  int b0    = group * 16;
  int m     = lane & 15;               // A-matrix row owned by this lane
  int t2    = lane >> 4;               // K half: lanes 16-31 hold K=2,3

  const float* xrow = x + (size_t)(b0 + m) * Fn;
  v8f acc0 = {}; v8f acc1 = {};

  for (int q = 0; q < Fn / 4; ++q) {   // 4 features (2 pairs) per iteration
    float4 xq = *(const float4*)(xrow + q * 4);
#pragma unroll
    for (int pp = 0; pp < 2; ++pp) {
      float x0 = pp ? xq.z : xq.x;
      float x1 = pp ? xq.w : xq.y;
      float xm = t2 ? x1 : x0;         // mixed chunk: half-wave per feature
      int base = (q * 2 + pp) * PR + t2 * 2;
#pragma unroll
      for (int ch = 0; ch < 5; ++ch) {
        int o = base + ch * 4;
        float xv = (ch < 2) ? x0 : (ch < 4) ? x1 : xm;
        v2f A, Bv;
        A.x  = fast_tanh(fmaf(xv, sW1[o],     sB1[o]));
        A.y  = fast_tanh(fmaf(xv, sW1[o + 1], sB1[o + 1]));
        Bv.x = sC[o];
        Bv.y = sC[o + 1];
        if ((ch & 1) == 0)
          acc0 = __builtin_amdgcn_wmma_f32_16x16x4_f32(
                     false, A, false, Bv, (short)0, acc0, false, false);
        else
          acc1 = __builtin_amdgcn_wmma_f32_16x16x4_f32(
                     false, A, false, Bv, (short)0, acc1, false, false);
      }
    }
  }

  v8f accT = acc0 + acc1;
  // D layout: VGPR r / lane 0 -> (M=r, N=0); VGPR r / lane 16 -> (M=8+r, N=0)
  if (lane == 0) {
#pragma unroll
    for (int r = 0; r < 8; ++r) y[b0 + r] = accT[r] + c0;
  } else if (lane == 16) {
#pragma unroll
    for (int r = 0; r < 8; ++r) y[b0 + 8 + r] = accT[r] + c0;
  }
}

extern "C" void kernel_launch(void* const* d_in, const int* in_sizes, int n_in,
                              void* d_out, int out_size, void* d_ws, size_t ws_size,
                              hipStream_t stream) {
  const float* x     = (const float*)d_in[0];
  const float* W1    = (const float*)d_in[1];
  const float* b1    = (const float*)d_in[2];
  const float* W2    = (const float*)d_in[3];
  const float* b2    = (const float*)d_in[4];
  const float* theta = (const float*)d_in[5];
  const float* bias  = (const float*)d_in[6];

  float* yout      = (float*)d_out;        // y_pred: [0, 32768)
  float* out_theta = yout + Bn;            // nn_theta: [32768, 32768+512)

  float* ws_theta = (float*)d_ws;          // 512 floats
  float* ws_const = ws_theta + Fn;         // 1 float

  nul_prep<<<1, Fn, 0, stream>>>(theta, b2, bias, ws_theta, ws_const, out_theta);
  nul_main<<<512, 128, 0, stream>>>(x, W1, b1, W2, ws_theta, ws_const, yout);
}